// smooth_loss_88983132439214
// MI455X (gfx1250) — compile-verified
//
#include <hip/hip_runtime.h>

// ---------------------------------------------------------------------------
// reg = (1/N) [ sum_i (1 - 1/deg_i) ||H_i||^2
//               - sum_e <H_row, H_col> * dinv[row] * dinv[col] ]
// deg_i = (#edges with row==i) + 1,  dinv = deg^{-1/2}
//
// Phase 3 (edge dots) uses V_WMMA_F32_16X16X4_F32: 16 edges/wave form a
// 16x16 Gram tile over K=64 (16 chained K=4 WMMAs, fp32 accumulate); we keep
// only the diagonal. Same memory traffic as a VALU dot, full fp32 precision.
// ---------------------------------------------------------------------------

typedef __attribute__((ext_vector_type(2))) float v2f;
typedef __attribute__((ext_vector_type(4))) float v4f;
typedef __attribute__((ext_vector_type(8))) float v8f;

#define DDIM 64  // hidden feature dim (fixed by the graph pool)

// ---- zero deg counters + scalar accumulator (ws is poisoned, not zeroed) ---
__global__ void zero_ws_kernel(unsigned int* __restrict__ deg,
                               float* __restrict__ accum, int n) {
  int i = blockIdx.x * blockDim.x + threadIdx.x;
  int stride = gridDim.x * blockDim.x;
  for (int k = i; k < n; k += stride) deg[k] = 0u;
  if (i == 0) *accum = 0.0f;
}

// ---- deg[i] = #edges with row==i (atomics into L2-resident counters) ------
__global__ void deg_kernel(const int* __restrict__ row,
                           unsigned int* __restrict__ deg, int e) {
  int i = blockIdx.x * blockDim.x + threadIdx.x;
  int stride = gridDim.x * blockDim.x;
  for (int k = i; k < e; k += stride) atomicAdd(&deg[row[k]], 1u);
}

// ---- node term: (1 - 1/deg_i) * ||H_i||^2 ; also emit dinv[i] -------------
// 16 threads per node, each loads one float4 (coalesced 256B rows).
__global__ void node_kernel(const float* __restrict__ H,
                            const unsigned int* __restrict__ deg,
                            float* __restrict__ dinv,
                            float* __restrict__ accum, int n) {
  const int tid = blockIdx.x * blockDim.x + threadIdx.x;
  const int grp = tid >> 4;
  const int sub = tid & 15;
  const int ngroups = (gridDim.x * blockDim.x) >> 4;
  float partial = 0.0f;

  for (int node = grp; node < n; node += ngroups) {
    v4f h = *(const v4f*)(H + (size_t)node * DDIM + sub * 4);
    float s = h.x * h.x + h.y * h.y + h.z * h.z + h.w * h.w;
    // reduce across the 16-lane group (xor masks < 16 stay in-group)
    s += __shfl_xor(s, 1, 32);
    s += __shfl_xor(s, 2, 32);
    s += __shfl_xor(s, 4, 32);
    s += __shfl_xor(s, 8, 32);
    if (sub == 0) {
      float dg = (float)deg[node] + 1.0f;  // + A_COEF self-loop
      float di = rsqrtf(dg);
      dinv[node] = di;
      partial += (1.0f - 1.0f / dg) * s;
    }
  }
  // fold the two group leaders (lanes 0 and 16) of each wave; others hold 0
  partial += __shfl_xor(partial, 16, 32);
  partial += __shfl_xor(partial, 8, 32);
  partial += __shfl_xor(partial, 4, 32);
  partial += __shfl_xor(partial, 2, 32);
  partial += __shfl_xor(partial, 1, 32);
  if ((threadIdx.x & 31) == 0) atomicAdd(accum, partial);
}

// ---- edge term via f32 WMMA Gram-diagonal ---------------------------------
// Tile = 16 edges per wave iteration. Operand layout for
// V_WMMA_F32_16X16X4_F32 (ISA 7.12.2): A lane m holds row m, K striped as
// {2*half, 2*half+1} across the v2f pair; B lane n holds column n likewise.
// So every lane loads one contiguous float2 per K-step from its edge's
// row-node (A) / col-node (B) feature vector.
__global__ void edge_kernel(const int* __restrict__ row,
                            const int* __restrict__ col,
                            const float* __restrict__ H,
                            const float* __restrict__ dinv,
                            float* __restrict__ accum, int e) {
  const int lane = threadIdx.x & 31;
  const int sub = lane & 15;          // which edge of the tile this lane owns
  const int half = lane >> 4;         // K offset selector
  const int wavesPerBlk = blockDim.x >> 5;
  const int waveId = blockIdx.x * wavesPerBlk + (threadIdx.x >> 5);
  const int numWaves = gridDim.x * wavesPerBlk;
  const int numTiles = (e + 15) >> 4;

  float partial = 0.0f;

  for (int tile = waveId; tile < numTiles; tile += numWaves) {
    const int eidx = tile * 16 + sub;
    const int ec = (eidx < e) ? eidx : (e - 1);  // clamp, keep EXEC full
    const int r = row[ec];
    const int c = col[ec];
    const float* ap = H + (size_t)r * DDIM + 2 * half;  // A: 16x64 rows
    const float* bp = H + (size_t)c * DDIM + 2 * half;  // B: 64x16 cols

    v8f acc = {};
#pragma unroll
    for (int j = 0; j < 16; ++j) {  // K = 64 in steps of 4
      v2f a = *(const v2f*)(ap + 4 * j);
      v2f b = *(const v2f*)(bp + 4 * j);
      acc = __builtin_amdgcn_wmma_f32_16x16x4_f32(
          /*neg_a=*/false, a, /*neg_b=*/false, b,
          /*c_mod=*/(short)0, acc, /*reuse_a=*/false, /*reuse_b=*/false);
    }

    // Diagonal of the 16x16 fp32 accumulator (ISA C/D layout):
    //   diag d, d<8  -> lane d,     VGPR d
    //   diag d, d>=8 -> lane d+16,  VGPR d-8
    // The owning lane's `sub` equals d, so its r/c are edge d's endpoints.
    const int sel = (lane < 8) ? lane : ((lane >= 24) ? (lane - 24) : -1);
    float dval = 0.0f;
#pragma unroll
    for (int k = 0; k < 8; ++k) dval = (sel == k) ? acc[k] : dval;

    const bool valid = (sel >= 0) && (eidx < e);
    const float coef = -dinv[r] * dinv[c];  // M2 * deg_r^-1/2 * deg_c^-1/2
    partial += valid ? dval * coef : 0.0f;
  }

  partial += __shfl_xor(partial, 16, 32);
  partial += __shfl_xor(partial, 8, 32);
  partial += __shfl_xor(partial, 4, 32);
  partial += __shfl_xor(partial, 2, 32);
  partial += __shfl_xor(partial, 1, 32);
  if (lane == 0) atomicAdd(accum, partial);
}

// ---- reg = accum / N ------------------------------------------------------
__global__ void finalize_kernel(const float* __restrict__ accum,
                                float* __restrict__ out, float invN) {
  if (blockIdx.x == 0 && threadIdx.x == 0) out[0] = accum[0] * invN;
}

extern "C" void kernel_launch(void* const* d_in, const int* in_sizes, int n_in,
                              void* d_out, int out_size, void* d_ws,
                              size_t ws_size, hipStream_t stream) {
  const int* edge = (const int*)d_in[0];    // [2, E] int32 (JAX x64 off)
  const float* H = (const float*)d_in[1];   // [N, 64] float32
  const int E_ = in_sizes[0] / 2;
  const int N_ = in_sizes[1] / DDIM;
  const int* row = edge;
  const int* col = edge + E_;

  // ws layout: u32 deg[N] | f32 dinv[N] | f32 accum
  unsigned int* deg = (unsigned int*)d_ws;
  float* dinv = (float*)((char*)d_ws + (size_t)N_ * sizeof(unsigned int));
  float* accum = (float*)((char*)d_ws + (size_t)N_ * 2u * sizeof(unsigned int));

  zero_ws_kernel<<<(N_ + 255) / 256, 256, 0, stream>>>(deg, accum, N_);
  deg_kernel<<<1024, 256, 0, stream>>>(row, deg, E_);
  node_kernel<<<1024, 256, 0, stream>>>(H, deg, dinv, accum, N_);
  edge_kernel<<<512, 256, 0, stream>>>(row, col, H, dinv, accum, E_);
  finalize_kernel<<<1, 1, 0, stream>>>(accum, (float*)d_out, 1.0f / (float)N_);
}